// GraphUnpool_70858370449711
// MI455X (gfx1250) — compile-verified
//
#include <hip/hip_runtime.h>

#define DIM 128
#define KSTEPS 32          // DIM / 4 (K=4 per f32 WMMA)
#define GEMM_WAVES 4       // waves (16-row blocks) per workgroup

typedef __attribute__((ext_vector_type(2))) float v2f;
typedef __attribute__((ext_vector_type(8))) float v8f;

#if defined(__HIP_DEVICE_COMPILE__) && __has_builtin(__builtin_amdgcn_wmma_f32_16x16x4_f32)
#define HAVE_WMMA_F32 1
#endif

// ---------------------------------------------------------------- zero fill
__global__ void zero_f32_kernel(float* __restrict__ p, long n) {
  long i = (long)blockIdx.x * blockDim.x + threadIdx.x;
  long stride = (long)gridDim.x * blockDim.x;
  for (; i < n; i += stride) p[i] = 0.0f;
}

// ---------------------------------------------------------------- degrees
__global__ void degree_kernel(const int* __restrict__ src, const int* __restrict__ dst,
                              float* __restrict__ deg_out, float* __restrict__ deg_in,
                              int ne) {
  int e = blockIdx.x * blockDim.x + threadIdx.x;
  if (e < ne) {
    atomicAdd(&deg_out[src[e]], 1.0f);
    atomicAdd(&deg_in[dst[e]], 1.0f);
  }
}

// ---------------------------------------------------------------- GEMM (+scatter into fine rows)
// One wave32 computes a 16x128 tile:  C = (feat[16 rows] * deg_out^-1/2) @ W
// then scatters the 16 rows to h_fine[select_idx[row]].
__global__ void __launch_bounds__(GEMM_WAVES * 32)
gemm_scatter_kernel(const float* __restrict__ feat, const int* __restrict__ sel,
                    const float* __restrict__ deg_out, const float* __restrict__ weight,
                    float* __restrict__ hfine, int nc) {
  const int lane = threadIdx.x & 31;
  const int wave = threadIdx.x >> 5;
  const int row0 = (blockIdx.x * GEMM_WAVES + wave) * 16;
  if (row0 >= nc) return;                       // uniform per wave
#if defined(HAVE_WMMA_F32)
  const int m  = lane & 15;                     // A: row index / B,C: column index
  const int kh = lane >> 4;                     // K-pair half select

  // --- A-operand role of this lane: row (row0+m), K elements {4s+2kh, 4s+2kh+1}
  int  cr  = row0 + m;
  bool ok  = cr < nc;
  int  crc = ok ? cr : 0;
  float scale = ok ? rsqrtf(fmaxf(deg_out[sel[crc]], 1.0f)) : 0.0f;

  v2f a[KSTEPS];
  const float* ap = feat + (long)crc * DIM + 2 * kh;
  #pragma unroll
  for (int s = 0; s < KSTEPS; ++s) {
    v2f av = *(const v2f*)(ap + 4 * s);         // 8B-aligned pair
    a[s] = av * scale;
  }

  // --- store targets: C VGPR i holds row (8*kh + i) of the tile
  int  strow[8];
  bool stok[8];
  #pragma unroll
  for (int i = 0; i < 8; ++i) {
    int rr   = row0 + 8 * kh + i;
    stok[i]  = rr < nc;
    strow[i] = sel[stok[i] ? rr : 0];
  }

  for (int nt = 0; nt < 8; ++nt) {              // 8 column tiles of 16
    const float* wp = weight + nt * 16 + m;     // B: lane = column n = nt*16 + m
    v8f acc = {};
    #pragma unroll
    for (int s = 0; s < KSTEPS; ++s) {
      int k = 4 * s + 2 * kh;
      v2f b;
      b[0] = wp[(long)k * DIM];
      b[1] = wp[(long)(k + 1) * DIM];
      acc = __builtin_amdgcn_wmma_f32_16x16x4_f32(false, a[s], false, b,
                                                  (short)0, acc, false, false);
    }
    #pragma unroll
    for (int i = 0; i < 8; ++i) {
      if (stok[i]) hfine[(long)strow[i] * DIM + nt * 16 + m] = acc[i];
    }
  }
#else
  // Scalar fallback (host pass / builtin unavailable): same launch geometry.
  for (int t = lane; t < 16 * DIM; t += 32) {
    int mi = t >> 7, n = t & (DIM - 1);
    int cr = row0 + mi;
    if (cr < nc) {
      int fr = sel[cr];
      float scale = rsqrtf(fmaxf(deg_out[fr], 1.0f));
      const float* ar = feat + (long)cr * DIM;
      float acc = 0.0f;
      for (int k = 0; k < DIM; ++k) acc += ar[k] * scale * weight[(long)k * DIM + n];
      hfine[(long)fr * DIM + n] = acc;
    }
  }
#endif
}

// ---------------------------------------------------------------- SpMM edge scatter
// One wave per edge: wave-uniform src/dst, lanes cover the 128 cols as float4.
// Rows that were never selected are all-zero -> whole wave skips the atomics.
__global__ void edge_scatter_kernel(const int* __restrict__ src, const int* __restrict__ dst,
                                    const float* __restrict__ hfine, float* __restrict__ out,
                                    int ne) {
  long t = (long)blockIdx.x * blockDim.x + threadIdx.x;
  int  e = (int)(t >> 5);
  if (e >= ne) return;
  int c = ((int)t & 31) << 2;                   // 4 floats per lane
  int s = src[e];                               // wave-uniform -> scalar load
  const float4 v = *(const float4*)(hfine + (long)s * DIM + c);
  if (v.x != 0.0f || v.y != 0.0f || v.z != 0.0f || v.w != 0.0f) {
    float* o = out + (long)dst[e] * DIM + c;
    atomicAdd(o + 0, v.x);
    atomicAdd(o + 1, v.y);
    atomicAdd(o + 2, v.z);
    atomicAdd(o + 3, v.w);
  }
}

// ---------------------------------------------------------------- finalize: *deg_in^-1/2 + bias
__global__ void finalize_kernel(float* __restrict__ out, const float* __restrict__ deg_in,
                                const float* __restrict__ bias, int nf) {
  int i = blockIdx.x * blockDim.x + threadIdx.x;
  if (i < nf * DIM) {
    int r = i >> 7, c = i & (DIM - 1);
    float nrm = rsqrtf(fmaxf(deg_in[r], 1.0f));
    out[i] = out[i] * nrm + bias[c];
  }
}

// ---------------------------------------------------------------- launch
extern "C" void kernel_launch(void* const* d_in, const int* in_sizes, int n_in,
                              void* d_out, int out_size, void* d_ws, size_t ws_size,
                              hipStream_t stream) {
  const float* feat   = (const float*)d_in[0];
  const int*   src    = (const int*)d_in[1];
  const int*   dst    = (const int*)d_in[2];
  const int*   sel    = (const int*)d_in[3];
  const float* weight = (const float*)d_in[4];
  const float* bias   = (const float*)d_in[5];

  const int ne = in_sizes[1];
  const int nc = in_sizes[3];
  const int nf = out_size / DIM;               // N_FINE = out rows

  float* hfine   = (float*)d_ws;               // nf*DIM floats (25.6 MB)
  float* deg_out = hfine + (size_t)nf * DIM;   // nf floats
  float* deg_in  = deg_out + nf;               // nf floats
  float* out     = (float*)d_out;

  const long ws_floats = (long)nf * DIM + 2L * nf;

  zero_f32_kernel<<<2048, 256, 0, stream>>>(hfine, ws_floats);
  zero_f32_kernel<<<2048, 256, 0, stream>>>(out, (long)nf * DIM);

  degree_kernel<<<(ne + 255) / 256, 256, 0, stream>>>(src, dst, deg_out, deg_in, ne);

  int nblk16 = (nc + 15) / 16;
  int ggrid  = (nblk16 + GEMM_WAVES - 1) / GEMM_WAVES;
  gemm_scatter_kernel<<<ggrid, GEMM_WAVES * 32, 0, stream>>>(feat, sel, deg_out, weight,
                                                             hfine, nc);

  long  etotal = (long)ne * 32;
  int   egrid  = (int)((etotal + 255) / 256);
  edge_scatter_kernel<<<egrid, 256, 0, stream>>>(src, dst, hfine, out, ne);

  finalize_kernel<<<(nf * DIM + 255) / 256, 256, 0, stream>>>(out, deg_in, bias, nf);
}